// Operator_18408229830853
// MI455X (gfx1250) — compile-verified
//
#include <hip/hip_runtime.h>
#include <hip/hip_bf16.h>

#define PI_F 3.14159265358979323846f

constexpr int Bsz = 2, Tt = 12, Ll = 3, Hh = 360, Ww = 720;
constexpr int TH = 5;                                   // T//2 - 1 POD modes
constexpr long long Xlen = (long long)Ll * Hh * Ww * 2; // 1,555,200
// workspace layout (float offsets)
constexpr long long XF_OFF = 0;
constexpr long long XF_SZ  = (long long)Bsz * TH * Xlen; // 15,552,000 floats (62.2 MB)
constexpr long long DX_OFF = XF_OFF + XF_SZ;
constexpr long long Y_OFF  = DX_OFF + Hh;
constexpr long long CA_OFF = Y_OFF + Hh;
constexpr long long CB_OFF = CA_OFF + Hh;
constexpr long long CC_OFF = CB_OFF + Hh;
constexpr long long G_OFF  = CC_OFF + Hh;                       // Bsz*25 gram
constexpr long long GV_OFF = G_OFF + (long long)Bsz * TH * TH;  // Bsz*5 g-vector
constexpr long long DIF_OFF = GV_OFF + (long long)Bsz * TH;     // L diffusivities

typedef float v2f __attribute__((ext_vector_type(2)));
typedef float v8f __attribute__((ext_vector_type(8)));

// ---------------- stencil helpers ----------------
__device__ __forceinline__ float gradW(const float* row, int w) {
    if (w == 0)      return row[1] - row[0];
    if (w == Ww - 1) return row[Ww - 1] - row[Ww - 2];
    return 0.5f * (row[w + 1] - row[w - 1]);
}

__device__ __forceinline__ float cy_at(const float* plane, int h, int w,
                                       const float* cA, const float* cB, const float* cC) {
    int hm = h > 0 ? h - 1 : 0;
    int hp = h < Hh - 1 ? h + 1 : Hh - 1;
    return cA[h] * plane[(size_t)hm * Ww + w] + cB[h] * plane[(size_t)h * Ww + w] +
           cC[h] * plane[(size_t)hp * Ww + w];
}

__device__ __forceinline__ void derivs(const float* plane, int h, int w, float dxh,
                                       const float* cA, const float* cB, const float* cC,
                                       float D, float& cx, float& cy, float& divt) {
    const float* row = plane + (size_t)h * Ww;
    cx = gradW(row, w) / dxh;
    cy = cy_at(plane, h, w, cA, cB, cC);
    float t1;
    if (w == 0)           t1 = gradW(row, 1) / dxh - cx;
    else if (w == Ww - 1) t1 = cx - gradW(row, Ww - 2) / dxh;
    else                  t1 = 0.5f * (gradW(row, w + 1) / dxh - gradW(row, w - 1) / dxh);
    float divx = t1 / dxh;
    int hm = h > 0 ? h - 1 : 0;
    int hp = h < Hh - 1 ? h + 1 : Hh - 1;
    float cym = cy_at(plane, hm, w, cA, cB, cC);
    float cyp = cy_at(plane, hp, w, cA, cB, cC);
    float divy = cA[h] * cym + cB[h] * cy + cC[h] * cyp;
    divt = D * (divx + divy);
}

// ---------------- kernel 0: geometry setup ----------------
__global__ void setup_kernel(const float* __restrict__ Rr, const float* __restrict__ diffu,
                             float* __restrict__ ws) {
    float* dxA = ws + DX_OFF;
    float* yA  = ws + Y_OFF;
    float* cA  = ws + CA_OFF;
    float* cB  = ws + CB_OFF;
    float* cC  = ws + CC_OFF;
    int h = threadIdx.x;
    if (h < Hh) {
        float lat = PI_F * ((float)h * (360.0f / (float)(Hh - 1))) / 180.0f;
        dxA[h] = 2.0f * PI_F / (float)Hh * Rr[0] * cosf(lat);
    }
    __syncthreads();
    if (h == 0) {
        float s = 0.0f;
        for (int i = 0; i < Hh; ++i) { s += dxA[i]; yA[i] = s; }
    }
    __syncthreads();
    if (h < Hh) {
        if (h == 0) {
            float dd = yA[1] - yA[0];
            cA[0] = 0.0f; cB[0] = -1.0f / dd; cC[0] = 1.0f / dd;
        } else if (h == Hh - 1) {
            float dd = yA[Hh - 1] - yA[Hh - 2];
            cA[h] = -1.0f / dd; cB[h] = 1.0f / dd; cC[h] = 0.0f;
        } else {
            float hl = yA[h] - yA[h - 1], hr = yA[h + 1] - yA[h];
            float den = hl * hr * (hl + hr);
            cA[h] = -hr * hr / den;
            cB[h] = (hr * hr - hl * hl) / den;
            cC[h] = hl * hl / den;
        }
    }
    if (threadIdx.x < Bsz * TH * TH) ws[G_OFF + threadIdx.x] = 0.0f; // zero gram
    if (threadIdx.x < Ll) ws[DIF_OFF + threadIdx.x] = diffu[threadIdx.x];
}

// ---------------- kernel 1: velocities xf via 2x2 lstsq ----------------
__global__ __launch_bounds__(256) void xf_kernel(const float* __restrict__ c,
                                                 const float* __restrict__ dt,
                                                 const float* __restrict__ diffu,
                                                 float* __restrict__ ws) {
    long long idx = (long long)blockIdx.x * blockDim.x + threadIdx.x;
    const long long NP = (long long)Bsz * TH * Ll * Hh * Ww;
    if (idx >= NP) return;
    int w = (int)(idx % Ww);
    int h = (int)((idx / Ww) % Hh);
    int l = (int)((idx / ((long long)Ww * Hh)) % Ll);
    int p = (int)((idx / ((long long)Ww * Hh * Ll)) % TH);
    int b = (int)(idx / ((long long)Ww * Hh * Ll * TH));

    const float* dxA = ws + DX_OFF;
    const float* cA  = ws + CA_OFF;
    const float* cB  = ws + CB_OFF;
    const float* cC  = ws + CC_OFF;
    float dxh = dxA[h];
    float D   = diffu[l];
    float dt0 = dt[0];

    int t0 = 2 * p;
    const size_t planeSz = (size_t)Hh * Ww;
    const float* pl0 = c + (((size_t)b * Tt + t0) * Ll + l) * planeSz;
    const float* pl1 = pl0 + (size_t)Ll * planeSz;  // t0+1
    const float* pl2 = pl1 + (size_t)Ll * planeSz;  // t0+2

    float cx0, cy0, dv0, cx1, cy1, dv1;
    derivs(pl0, h, w, dxh, cA, cB, cC, D, cx0, cy0, dv0);
    derivs(pl1, h, w, dxh, cA, cB, cC, D, cx1, cy1, dv1);

    float cc0 = pl0[(size_t)h * Ww + w];
    float cc1 = pl1[(size_t)h * Ww + w];
    float cc2 = pl2[(size_t)h * Ww + w];
    float a0 = (cc1 - cc0) / dt0 - dv0;
    float a1 = (cc2 - cc1) / dt0 - dv1;

    float det = cx0 * cy1 - cy0 * cx1;
    float v0 = (cy1 * a0 - cy0 * a1) / det;
    float v1 = (cx0 * a1 - cx1 * a0) / det;

    float* xf = ws + XF_OFF;
    long long x0 = ((long long)(l * Hh + h) * Ww + w) * 2;
    xf[((long long)b * TH + p) * Xlen + x0]     = v0;
    xf[((long long)b * TH + p) * Xlen + x0 + 1] = v1;
}

// ---------------- kernel 2: 5x5 Gram via V_WMMA_F32_16X16X4_F32 ----------------
// G[b][i][j] = sum_X xf[b,i,X]*xf[b,j,X].  A[m][k]=xf[m,X0+k], B[k][n]=xf[n,X0+k]
// hold identical data -> pass the same registers (layout-permutation invariant).
// Lanes with mode >= 5 load CLAMPED (valid) data: their products only land in
// C rows/cols >= 5, which are never read -> no predication in the hot loop.
constexpr int WPB = 1080; // waves per batch element
constexpr int GU  = 4;    // unroll (dual accumulators)
__global__ __launch_bounds__(256) void gram_kernel(const float* __restrict__ xf,
                                                   float* __restrict__ G) {
    int wave = (int)((blockIdx.x * blockDim.x + threadIdx.x) >> 5);
    int lane = threadIdx.x & 31;
    int b  = wave / WPB;
    int wv = wave % WPB;
    int m    = lane & 15;
    int half = lane >> 4;
    int mm   = (m < TH) ? m : (TH - 1); // clamp: junk stays in unread C rows/cols

    const float* pb = xf + (long long)b * TH * Xlen + (long long)mm * Xlen + 2 * half;
    v8f acc0 = {}, acc1 = {};
    const int NC  = (int)(Xlen / 4);     // 388,800 K-chunks
    const int NIT = NC / (WPB * GU);     // 90 exact (1080*4*90 == 388800)
    for (int it = 0; it < NIT; ++it) {
        long long k0 = (long long)wv + (long long)it * (GU * WPB);
        v2f a0 = *reinterpret_cast<const v2f*>(pb + 4 * (k0 + 0LL * WPB));
        v2f a1 = *reinterpret_cast<const v2f*>(pb + 4 * (k0 + 1LL * WPB));
        v2f a2 = *reinterpret_cast<const v2f*>(pb + 4 * (k0 + 2LL * WPB));
        v2f a3 = *reinterpret_cast<const v2f*>(pb + 4 * (k0 + 3LL * WPB));
        acc0 = __builtin_amdgcn_wmma_f32_16x16x4_f32(false, a0, false, a0,
                                                     (short)0, acc0, false, false);
        acc1 = __builtin_amdgcn_wmma_f32_16x16x4_f32(false, a1, false, a1,
                                                     (short)0, acc1, false, false);
        acc0 = __builtin_amdgcn_wmma_f32_16x16x4_f32(false, a2, false, a2,
                                                     (short)0, acc0, false, false);
        acc1 = __builtin_amdgcn_wmma_f32_16x16x4_f32(false, a3, false, a3,
                                                     (short)0, acc1, false, false);
    }
    // D layout: VGPR r, lanes 0-15 -> (M=r, N=lane). Need rows/cols 0..4.
    if (half == 0 && m < TH) {
        for (int r = 0; r < TH; ++r)
            atomicAdd(&G[(b * TH + r) * TH + m], acc0[r] + acc1[r]);
    }
}

// ---------------- kernel 3: per-b 5x5 eigensolve + fold A_v -> g[b][t] ----------------
__global__ void solve_kernel(const float* __restrict__ G, const float* __restrict__ Av,
                             float* __restrict__ gvec) {
    if (threadIdx.x != 0) return;
    int b = blockIdx.x;
    double M[TH][TH], V[TH][TH];
    for (int i = 0; i < TH; ++i)
        for (int j = 0; j < TH; ++j)
            M[i][j] = 0.5 * ((double)G[(b * TH + i) * TH + j] + (double)G[(b * TH + j) * TH + i]);
    double Gd[TH][TH];
    for (int i = 0; i < TH; ++i)
        for (int j = 0; j < TH; ++j) Gd[i][j] = M[i][j];
    for (int i = 0; i < TH; ++i)
        for (int j = 0; j < TH; ++j) V[i][j] = (i == j) ? 1.0 : 0.0;
    // cyclic Jacobi
    for (int sweep = 0; sweep < 60; ++sweep) {
        double off = 0.0, frob = 0.0;
        for (int i = 0; i < TH; ++i)
            for (int j = 0; j < TH; ++j) {
                frob += M[i][j] * M[i][j];
                if (i < j) off += M[i][j] * M[i][j];
            }
        if (off <= 1e-28 * (frob + 1e-300)) break;
        for (int p = 0; p < TH - 1; ++p)
            for (int q = p + 1; q < TH; ++q) {
                double apq = M[p][q];
                if (fabs(apq) < 1e-300) continue;
                double theta = 0.5 * (M[q][q] - M[p][p]) / apq;
                double t = ((theta >= 0.0) ? 1.0 : -1.0) /
                           (fabs(theta) + sqrt(theta * theta + 1.0));
                double cth = 1.0 / sqrt(t * t + 1.0), sth = t * cth;
                for (int r = 0; r < TH; ++r) { // M = M*J
                    double mrp = M[r][p], mrq = M[r][q];
                    M[r][p] = cth * mrp - sth * mrq;
                    M[r][q] = sth * mrp + cth * mrq;
                }
                for (int r = 0; r < TH; ++r) { // M = J^T*M
                    double mpr = M[p][r], mqr = M[q][r];
                    M[p][r] = cth * mpr - sth * mqr;
                    M[q][r] = sth * mpr + cth * mqr;
                }
                for (int r = 0; r < TH; ++r) { // V = V*J
                    double vrp = V[r][p], vrq = V[r][q];
                    V[r][p] = cth * vrp - sth * vrq;
                    V[r][q] = sth * vrp + cth * vrq;
                }
            }
    }
    // sort eigenvalues descending
    double lam[TH]; int ord[TH];
    for (int i = 0; i < TH; ++i) { lam[i] = M[i][i]; ord[i] = i; }
    for (int i = 0; i < TH; ++i)
        for (int j = i + 1; j < TH; ++j)
            if (lam[ord[j]] > lam[ord[i]]) { int t = ord[i]; ord[i] = ord[j]; ord[j] = t; }
    // vt[m][t] = V[t][ord[m]], sign-canonicalized (max-|.| entry positive)
    double vt[TH][TH], d[TH];
    for (int mI = 0; mI < TH; ++mI) {
        int col = ord[mI];
        double mx = 0.0; int mt = 0;
        for (int t = 0; t < TH; ++t) {
            vt[mI][t] = V[t][col];
            if (fabs(vt[mI][t]) > mx) { mx = fabs(vt[mI][t]); mt = t; }
        }
        if (vt[mI][mt] < 0.0)
            for (int t = 0; t < TH; ++t) vt[mI][t] = -vt[mI][t];
        double e = lam[col] > 0.0 ? lam[col] : 0.0;
        double s = sqrt(e);
        d[mI] = s / (s * s + ((s > 1.0) ? 0.0 : 1.0));
    }
    // coef[M] = d[M] * sum_t vt[M][t] * G[4][t]
    double coef[TH];
    for (int Mi = 0; Mi < TH; ++Mi) {
        double acc = 0.0;
        for (int t = 0; t < TH; ++t) acc += vt[Mi][t] * Gd[TH - 1][t];
        coef[Mi] = d[Mi] * acc;
    }
    // w[m] = sum_M coef[M]*A_v[M][m];  g[t] = sum_m w[m]*d[m]*vt[m][t]
    double wv[TH];
    for (int mI = 0; mI < TH; ++mI) {
        double acc = 0.0;
        for (int Mi = 0; Mi < TH; ++Mi) acc += coef[Mi] * (double)Av[Mi * TH + mI];
        wv[mI] = acc;
    }
    for (int t = 0; t < TH; ++t) {
        double acc = 0.0;
        for (int mI = 0; mI < TH; ++mI) acc += wv[mI] * d[mI] * vt[mI][t];
        gvec[b * TH + t] = (float)acc;
    }
}

// ---------------- kernel 4: final update at t = T-1 ----------------
// ylast[X] = xf[4,X] + sum_t g[t]*xf[t,X];  out = c + (div - ylast.(cx,cy))*dt
__global__ __launch_bounds__(256) void final_kernel(const float* __restrict__ c,
                                                    const float* __restrict__ dt,
                                                    const float* __restrict__ ws,
                                                    float* __restrict__ out) {
    long long idx = (long long)blockIdx.x * blockDim.x + threadIdx.x;
    const long long NP = (long long)Bsz * Ll * Hh * Ww;
    if (idx >= NP) return;
    int w = (int)(idx % Ww);
    int h = (int)((idx / Ww) % Hh);
    int l = (int)((idx / ((long long)Ww * Hh)) % Ll);
    int b = (int)(idx / ((long long)Ww * Hh * Ll));

    const float* dxA  = ws + DX_OFF;
    const float* cA   = ws + CA_OFF;
    const float* cB   = ws + CB_OFF;
    const float* cC   = ws + CC_OFF;
    const float* xf   = ws + XF_OFF;
    const float* gvec = ws + GV_OFF;
    float dxh = dxA[h];
    float dt0 = dt[0];
    float D   = ws[DIF_OFF + l];

    const size_t planeSz = (size_t)Hh * Ww;
    const float* pl = c + (((size_t)b * Tt + (Tt - 1)) * Ll + l) * planeSz;

    float cx, cy, divt;
    derivs(pl, h, w, dxh, cA, cB, cC, D, cx, cy, divt);

    long long x0 = ((long long)(l * Hh + h) * Ww + w) * 2;
    const float* xfb = xf + (long long)b * TH * Xlen;
    float y0 = xfb[(long long)(TH - 1) * Xlen + x0];
    float y1 = xfb[(long long)(TH - 1) * Xlen + x0 + 1];
    for (int t = 0; t < TH; ++t) {
        float gt = gvec[b * TH + t];
        y0 += gt * xfb[(long long)t * Xlen + x0];
        y1 += gt * xfb[(long long)t * Xlen + x0 + 1];
    }
    float adv = y0 * cx + y1 * cy;
    out[idx] = pl[(size_t)h * Ww + w] + (divt - adv) * dt0;
}

extern "C" void kernel_launch(void* const* d_in, const int* in_sizes, int n_in,
                              void* d_out, int out_size, void* d_ws, size_t ws_size,
                              hipStream_t stream) {
    (void)in_sizes; (void)n_in; (void)out_size; (void)ws_size;
    const float* c     = (const float*)d_in[0];
    const float* Rr    = (const float*)d_in[1];
    const float* dt    = (const float*)d_in[2];
    const float* diffu = (const float*)d_in[3];
    const float* Av    = (const float*)d_in[4];
    float* wsf = (float*)d_ws;

    setup_kernel<<<1, 512, 0, stream>>>(Rr, diffu, wsf);

    const long long N1 = (long long)Bsz * TH * Ll * Hh * Ww; // 7,776,000
    xf_kernel<<<(int)((N1 + 255) / 256), 256, 0, stream>>>(c, dt, diffu, wsf);

    const int gramBlocks = (Bsz * WPB) / 8; // 270 blocks x 8 waves
    gram_kernel<<<gramBlocks, 256, 0, stream>>>(wsf + XF_OFF, wsf + G_OFF);

    solve_kernel<<<Bsz, 32, 0, stream>>>(wsf + G_OFF, Av, wsf + GV_OFF);

    const long long N5 = (long long)Bsz * Ll * Hh * Ww; // 1,555,200
    final_kernel<<<(int)((N5 + 255) / 256), 256, 0, stream>>>(c, dt, wsf, (float*)d_out);
}